// MechanisticNRTLLoss_57123065037608
// MI455X (gfx1250) — compile-verified
//
#include <hip/hip_runtime.h>
#include <stdint.h>

// ---- constants from the reference ----
#define ALPHA_C   0.3f
#define RGAS_C    8.314462618f
#define EPS_C     1e-12f
#define TAUCLIP_C 10.0f
#define LNCLIP_C  20.0f
#define EPSFD_C   1e-4f
// LAM_PHY = 1.0, LAM_GD = 0.1, LAM_TPD = 0.1, MARGIN = 0.0

#define NBLK 1024
#define NTHR 128              // 4 waves (wave32)
#define TILE 128              // elements per LDS tile

// LDS tile layout (float words), per buffer
#define P_PRED 0              // 128*6 = 768
#define P_TGT  768            // 768
#define P_T    1536           // 128
#define P_G    1664           // 128*9 = 1152
#define P_DIR  2816           // 2*128*3 = 768
#define P_NOI  3584           // 4*128*3 = 1536
#define BUF_WORDS 5120        // 20 KB per buffer
// async b32 instructions issued per wave per tile:
// 768/128 + 768/128 + 128/128 + 1152/128 + 2*(384/128) + 4*(384/128) = 6+6+1+9+6+12
#define ASYNC_OPS 40

typedef __attribute__((address_space(1))) int as1_i;   // global int*
typedef __attribute__((address_space(3))) int as3_i;   // LDS int*

// ---- gfx1250 async global->LDS copy (ASYNCcnt-tracked, no VGPR returns) ----
__device__ __forceinline__ void async_b32(float* lds_dst, const float* gsrc) {
#if __has_builtin(__builtin_amdgcn_global_load_async_to_lds_b32)
    __builtin_amdgcn_global_load_async_to_lds_b32(
        (as1_i*)(uintptr_t)gsrc,
        (as3_i*)(uint32_t)(uintptr_t)lds_dst, 0, 0);
#else
    uint32_t loff = (uint32_t)(uintptr_t)lds_dst;   // low 32 bits = LDS byte offset
    asm volatile("global_load_async_to_lds_b32 %0, %1, off"
                 :: "v"(loff), "v"(gsrc) : "memory");
#endif
}

__device__ __forceinline__ void wait_async_prev() {   // ASYNCcnt <= ASYNC_OPS
#if __has_builtin(__builtin_amdgcn_s_wait_asynccnt)
    __builtin_amdgcn_s_wait_asynccnt(ASYNC_OPS);
#else
    asm volatile("s_wait_asynccnt 40" ::: "memory");
#endif
}

__device__ __forceinline__ void wait_async_all() {    // ASYNCcnt == 0
#if __has_builtin(__builtin_amdgcn_s_wait_asynccnt)
    __builtin_amdgcn_s_wait_asynccnt(0);
#else
    asm volatile("s_wait_asynccnt 0" ::: "memory");
#endif
}

// ---- math helpers ----
__device__ __forceinline__ float frcp(float x) { return __builtin_amdgcn_rcpf(x); }

__device__ __forceinline__ void renorm3(float x[3]) {
    x[0] = fmaxf(x[0], 0.0f);
    x[1] = fmaxf(x[1], 0.0f);
    x[2] = fmaxf(x[2], 0.0f);
    float inv = frcp(fmaxf(x[0] + x[1] + x[2], EPS_C));
    x[0] *= inv; x[1] *= inv; x[2] *= inv;
}

// ln_gamma with per-element tau/G/tau*G precomputed (shared by all 10 calls).
__device__ __forceinline__ void ln_gamma(const float xin[3], const float tau[9],
                                         const float G[9], const float tG[9],
                                         float lg[3]) {
    const float x0 = fmaxf(xin[0], 0.0f);
    const float x1 = fmaxf(xin[1], 0.0f);
    const float x2 = fmaxf(xin[2], 0.0f);
    float ratio[3], rd[3];
#pragma unroll
    for (int i = 0; i < 3; ++i) {
        float d = fmaf(x0, G[i],  fmaf(x1, G[3 + i],  x2 * G[6 + i]));
        float a = fmaf(x0, tG[i], fmaf(x1, tG[3 + i], x2 * tG[6 + i]));
        d = fmaxf(d, EPS_C);
        rd[i]    = frcp(d);
        ratio[i] = a * rd[i];
    }
    const float xs[3] = {x0, x1, x2};
#pragma unroll
    for (int i = 0; i < 3; ++i) {
        float t2 = 0.0f;
#pragma unroll
        for (int j = 0; j < 3; ++j) {
            float w = xs[j] * G[i * 3 + j] * rd[j];
            t2 = fmaf(w, tau[i * 3 + j] - ratio[j], t2);
        }
        lg[i] = fminf(fmaxf(ratio[i] + t2, -LNCLIP_C), LNCLIP_C);
    }
}

// full per-element loss contribution from the 40 input floats
__device__ __forceinline__ void accumulate_element(
    const float pr[6], const float tgt[6], float Tv, const float gm[9],
    const float dv[6], const float nz[12],
    float& s_sup, float& s_phy, float& s_gd, float& s_tpd)
{
#pragma unroll
    for (int k = 0; k < 6; ++k) { float d = pr[k] - tgt[k]; s_sup = fmaf(d, d, s_sup); }

    float tau[9], G[9], tG[9];
    {
        const float Tc    = fmaxf(Tv, 1.0f);
        const float invRT = 1.0f / (RGAS_C * Tc);
#pragma unroll
        for (int k = 0; k < 9; ++k) {
            float t = gm[k] * invRT;
            t = fminf(fmaxf(t, -TAUCLIP_C), TAUCLIP_C);
            tau[k] = t;
            G[k]   = __expf(-ALPHA_C * t);
            tG[k]  = t * G[k];
        }
    }

    float xE[3] = {pr[0], pr[1], pr[2]};
    float xR[3] = {pr[3], pr[4], pr[5]};
    renorm3(xE); renorm3(xR);
    float lgE[3], lgR[3];
    ln_gamma(xE, tau, G, tG, lgE);
    ln_gamma(xR, tau, G, tG, lgR);
    float lnxE[3];
#pragma unroll
    for (int k = 0; k < 3; ++k) {
        lnxE[k] = __logf(fmaxf(xE[k], EPS_C));
        float r = lnxE[k] + lgE[k] - __logf(fmaxf(xR[k], EPS_C)) - lgR[k];
        s_phy = fmaf(r, r, s_phy);
    }

#pragma unroll
    for (int d = 0; d < 2; ++d) {
        float xp[3], xm[3];
#pragma unroll
        for (int c = 0; c < 3; ++c) {
            float dvv = dv[d * 3 + c];
            xp[c] = fmaf( EPSFD_C, dvv, xE[c]);
            xm[c] = fmaf(-EPSFD_C, dvv, xE[c]);
        }
        renorm3(xp); renorm3(xm);
        float lgp[3], lgm[3];
        ln_gamma(xp, tau, G, tG, lgp);
        ln_gamma(xm, tau, G, tG, lgm);
        float gd = 0.0f;
#pragma unroll
        for (int c = 0; c < 3; ++c)
            gd = fmaf(xE[c], (lgp[c] - lgm[c]) * (0.5f / EPSFD_C), gd);
        s_gd = fmaf(gd, gd, s_gd);
    }

#pragma unroll
    for (int t = 0; t < 4; ++t) {
        float w[3] = {xE[0] + nz[t * 3 + 0], xE[1] + nz[t * 3 + 1], xE[2] + nz[t * 3 + 2]};
        renorm3(w);
        float lgw[3];
        ln_gamma(w, tau, G, tG, lgw);
        float tpd = 0.0f;
#pragma unroll
        for (int c = 0; c < 3; ++c)
            tpd = fmaf(w[c], __logf(fmaxf(w[c], EPS_C)) + lgw[c] - lnxE[c] - lgE[c], tpd);
        s_tpd += fmaxf(-tpd, 0.0f);   // MARGIN = 0
    }
}

__device__ __forceinline__ void copy_region(float* ldst, const float* gsrc,
                                            int words, int tid) {
#pragma unroll 1
    for (int i = tid; i < words; i += NTHR)
        async_b32(ldst + i, gsrc + i);
}

__device__ __forceinline__ void issue_tile(
    float* buf, const float* pred, const float* target, const float* T,
    const float* g, const float* dirs, const float* noise, int e0, int n, int tid)
{
    copy_region(buf + P_PRED, pred   + (size_t)e0 * 6, TILE * 6, tid);
    copy_region(buf + P_TGT,  target + (size_t)e0 * 6, TILE * 6, tid);
    copy_region(buf + P_T,    T      + e0,             TILE,     tid);
    copy_region(buf + P_G,    g      + (size_t)e0 * 9, TILE * 9, tid);
    copy_region(buf + P_DIR,            dirs + (size_t)e0 * 3,          TILE * 3, tid);
    copy_region(buf + P_DIR + TILE * 3, dirs + ((size_t)n + e0) * 3,    TILE * 3, tid);
#pragma unroll
    for (int t = 0; t < 4; ++t)
        copy_region(buf + P_NOI + t * TILE * 3,
                    noise + ((size_t)t * n + e0) * 3, TILE * 3, tid);
}

__global__ __launch_bounds__(NTHR) void nrtl_loss_partials(
    const float* __restrict__ pred, const float* __restrict__ target,
    const float* __restrict__ T,    const float* __restrict__ g,
    const float* __restrict__ dirs, const float* __restrict__ noise,
    float* __restrict__ partials, int n)
{
    __shared__ float  smem[2 * BUF_WORDS];   // 40 KB double buffer
    __shared__ float4 sred[NTHR];            //  2 KB reduction

    const int tid = threadIdx.x;
    float s_sup = 0.0f, s_phy = 0.0f, s_gd = 0.0f, s_tpd = 0.0f;

    const int ntiles = n / TILE;
    int tile = blockIdx.x;
    int cur  = 0;

    if (tile < ntiles)
        issue_tile(smem, pred, target, T, g, dirs, noise, tile * TILE, n, tid);

    for (; tile < ntiles; tile += gridDim.x) {
        const int nxt = tile + gridDim.x;
        if (nxt < ntiles) {
            // prefetch next tile into the other buffer, then wait for current tile:
            // async completions are in issue order, so ASYNCcnt<=ASYNC_OPS means
            // everything except the just-issued batch has landed in LDS.
            issue_tile(smem + (cur ^ 1) * BUF_WORDS, pred, target, T, g, dirs, noise,
                       nxt * TILE, n, tid);
            wait_async_prev();
        } else {
            wait_async_all();
        }
        __syncthreads();   // cross-wave visibility of LDS writes

        const float* buf = smem + cur * BUF_WORDS;
        float pr[6], tg[6], gm[9], dv[6], nz[12];
#pragma unroll
        for (int k = 0; k < 6; ++k) { pr[k] = buf[P_PRED + tid * 6 + k];
                                      tg[k] = buf[P_TGT  + tid * 6 + k]; }
        const float Tv = buf[P_T + tid];
#pragma unroll
        for (int k = 0; k < 9; ++k) gm[k] = buf[P_G + tid * 9 + k];
#pragma unroll
        for (int d = 0; d < 2; ++d)
#pragma unroll
            for (int c = 0; c < 3; ++c)
                dv[d * 3 + c] = buf[P_DIR + d * TILE * 3 + tid * 3 + c];
#pragma unroll
        for (int t = 0; t < 4; ++t)
#pragma unroll
            for (int c = 0; c < 3; ++c)
                nz[t * 3 + c] = buf[P_NOI + t * TILE * 3 + tid * 3 + c];

        accumulate_element(pr, tg, Tv, gm, dv, nz, s_sup, s_phy, s_gd, s_tpd);

        __syncthreads();   // all waves done reading before buffer is overwritten
        cur ^= 1;
    }

    // ---- tail: n % TILE leftover elements via direct loads ----
    {
        const int e_full = ntiles * TILE;
        const int gsz    = gridDim.x * NTHR;
        for (int b = e_full + blockIdx.x * NTHR + tid; b < n; b += gsz) {
            float pr[6], tg[6], gm[9], dv[6], nz[12];
#pragma unroll
            for (int k = 0; k < 6; ++k) { pr[k] = pred[(size_t)b * 6 + k];
                                          tg[k] = target[(size_t)b * 6 + k]; }
            const float Tv = T[b];
#pragma unroll
            for (int k = 0; k < 9; ++k) gm[k] = g[(size_t)b * 9 + k];
#pragma unroll
            for (int d = 0; d < 2; ++d)
#pragma unroll
                for (int c = 0; c < 3; ++c)
                    dv[d * 3 + c] = dirs[((size_t)d * n + b) * 3 + c];
#pragma unroll
            for (int t = 0; t < 4; ++t)
#pragma unroll
                for (int c = 0; c < 3; ++c)
                    nz[t * 3 + c] = noise[((size_t)t * n + b) * 3 + c];
            accumulate_element(pr, tg, Tv, gm, dv, nz, s_sup, s_phy, s_gd, s_tpd);
        }
    }

    // ---- deterministic block tree reduction ----
    sred[tid] = make_float4(s_sup, s_phy, s_gd, s_tpd);
    __syncthreads();
#pragma unroll
    for (int off = NTHR / 2; off > 0; off >>= 1) {
        if (tid < off) {
            float4 a = sred[tid], c = sred[tid + off];
            sred[tid] = make_float4(a.x + c.x, a.y + c.y, a.z + c.z, a.w + c.w);
        }
        __syncthreads();
    }
    if (tid == 0) {
        float4 a = sred[0];
        partials[(size_t)blockIdx.x * 4 + 0] = a.x;
        partials[(size_t)blockIdx.x * 4 + 1] = a.y;
        partials[(size_t)blockIdx.x * 4 + 2] = a.z;
        partials[(size_t)blockIdx.x * 4 + 3] = a.w;
    }
}

__global__ __launch_bounds__(NTHR) void nrtl_loss_finalize(
    const float* __restrict__ partials, int nblocks,
    float* __restrict__ out, float inv_n)
{
    float4 s = make_float4(0.f, 0.f, 0.f, 0.f);
    for (int i = threadIdx.x; i < nblocks; i += NTHR) {
        s.x += partials[(size_t)i * 4 + 0];
        s.y += partials[(size_t)i * 4 + 1];
        s.z += partials[(size_t)i * 4 + 2];
        s.w += partials[(size_t)i * 4 + 3];
    }
    __shared__ float4 red[NTHR];
    red[threadIdx.x] = s;
    __syncthreads();
#pragma unroll
    for (int off = NTHR / 2; off > 0; off >>= 1) {
        if ((int)threadIdx.x < off) {
            float4 a = red[threadIdx.x], c = red[threadIdx.x + off];
            red[threadIdx.x] = make_float4(a.x + c.x, a.y + c.y, a.z + c.z, a.w + c.w);
        }
        __syncthreads();
    }
    if (threadIdx.x == 0) {
        float4 a = red[0];
        float L = a.x * (inv_n * (1.0f / 6.0f))
                + a.y * (inv_n * (1.0f / 3.0f))
                + 0.1f * a.z * (inv_n * 0.5f)
                + 0.1f * a.w * (inv_n * 0.25f);
        out[0] = L;
    }
}

extern "C" void kernel_launch(void* const* d_in, const int* in_sizes, int n_in,
                              void* d_out, int out_size, void* d_ws, size_t ws_size,
                              hipStream_t stream) {
    const float* pred   = (const float*)d_in[0];   // (B,6)
    const float* target = (const float*)d_in[1];   // (B,6)
    const float* T      = (const float*)d_in[2];   // (B,)
    const float* g      = (const float*)d_in[3];   // (B,3,3)
    const float* dirs   = (const float*)d_in[4];   // (2,B,3)
    const float* noise  = (const float*)d_in[5];   // (4,B,3)
    const int n = in_sizes[2];                     // B from T's element count

    float* partials = (float*)d_ws;                // NBLK*4 floats = 16 KB
    nrtl_loss_partials<<<NBLK, NTHR, 0, stream>>>(pred, target, T, g, dirs, noise,
                                                  partials, n);
    nrtl_loss_finalize<<<1, NTHR, 0, stream>>>(partials, NBLK, (float*)d_out,
                                               1.0f / (float)n);
}